// PFMOnlyModelNoLearnable_24575802867934
// MI455X (gfx1250) — compile-verified
//
#include <hip/hip_runtime.h>
#include <stdint.h>

// Problem constants (from reference)
#define NB      256          // neighbors per agent
#define TSTEPS  12
#define AGENTS  (64 * 512)   // B * A
#define WAVES_PER_BLOCK 8
#define DT_     0.1f
#define RADIUS_ 0.5f

#if defined(__has_builtin)
#  if __has_builtin(__builtin_amdgcn_global_load_async_to_lds_b128)
#    define HAVE_ASYNC_LDS 1
#  endif
#  if __has_builtin(__builtin_amdgcn_s_wait_asynccnt)
#    define HAVE_WAIT_ASYNC 1
#  endif
#endif

// b128 async payload type: <4 x i32>, with explicit address spaces
typedef int v4i __attribute__((ext_vector_type(4)));
typedef __attribute__((address_space(1))) v4i GlobalV4;
typedef __attribute__((address_space(3))) v4i LdsV4;

__global__ __launch_bounds__(256) void pfm_rollout_kernel(
    const float* __restrict__ history,    // (B, A, 8, 2)
    const float* __restrict__ neighbors,  // (B, A, 256, 2)
    const float* __restrict__ goal,       // (B, A, 2)
    float* __restrict__ out)              // (B, A, 12, 2)
{
  const int wave  = threadIdx.x >> 5;
  const int lane  = threadIdx.x & 31;
  const int agent = blockIdx.x * WAVES_PER_BLOCK + wave;

  // 8 agents * 256 neighbors * 8B = 16KB neighbor staging + output staging
  __shared__ float4 nbr_lds[WAVES_PER_BLOCK * (NB / 2)];
  __shared__ float  out_stage[WAVES_PER_BLOCK][TSTEPS * 2];

  const float4* nbr_g    = (const float4*)(neighbors + (size_t)agent * (NB * 2));
  float4*       lds_base = &nbr_lds[wave * (NB / 2)];

  float4 n[4];  // 8 neighbors per lane, register-resident for all 12 steps

#ifdef HAVE_ASYNC_LDS
  // CDNA5 async DMA: global -> LDS, tracked by ASYNCcnt. Wave-private region,
  // so no cross-wave barrier is needed, just s_wait_asynccnt.
  #pragma unroll
  for (int k = 0; k < 4; ++k) {
    const int idx = lane + 32 * k;
    __builtin_amdgcn_global_load_async_to_lds_b128(
        (GlobalV4*)(uintptr_t)(const void*)(nbr_g + idx),
        (LdsV4*)(lds_base + idx),
        /*offset=*/0, /*cpol=*/0);
  }
#ifdef HAVE_WAIT_ASYNC
  __builtin_amdgcn_s_wait_asynccnt(0);
#else
  asm volatile("s_wait_asynccnt 0" ::: "memory");
#endif
  asm volatile("" ::: "memory");  // compiler ordering fence before LDS reads
  #pragma unroll
  for (int k = 0; k < 4; ++k) n[k] = lds_base[lane + 32 * k];
#else
  // Fallback: direct coalesced global_load_b128 into registers
  #pragma unroll
  for (int k = 0; k < 4; ++k) n[k] = nbr_g[lane + 32 * k];
#endif

  // Per-agent scalar state (replicated across all 32 lanes of the wave)
  const float* h = history + (size_t)agent * 16;   // H=8, last two rows: idx 6,7
  float cx = h[14], cy = h[15];                    // cur0 = history[..,7,:]
  float px = cx + (cx - h[12]);                    // pred0 = cur0 + (cur0 - history[..,6,:])
  float py = cy + (cy - h[13]);
  const float gx = goal[(size_t)agent * 2 + 0];
  const float gy = goal[(size_t)agent * 2 + 1];
  const float c0x = cx, c0y = cy;

  float pre_x[TSTEPS], pre_y[TSTEPS];

  // ---- rollout scan: 12 sequential force steps ----
  #pragma unroll
  for (int t = 0; t < TSTEPS; ++t) {
    float frx = 0.f, fry = 0.f;
    #pragma unroll
    for (int k = 0; k < 4; ++k) {
      {
        float dx = cx - n[k].x, dy = cy - n[k].y;
        float dist = sqrtf(fmaf(dx, dx, dy * dy)) + 1e-6f;
        float w = (dist < RADIUS_) ? (1.0f / (dist * dist)) : 0.0f;
        frx = fmaf(dx, w, frx);
        fry = fmaf(dy, w, fry);
      }
      {
        float dx = cx - n[k].z, dy = cy - n[k].w;
        float dist = sqrtf(fmaf(dx, dx, dy * dy)) + 1e-6f;
        float w = (dist < RADIUS_) ? (1.0f / (dist * dist)) : 0.0f;
        frx = fmaf(dx, w, frx);
        fry = fmaf(dy, w, fry);
      }
    }
    // wave32 butterfly reduction: every lane ends with the full neighbor sum
    #pragma unroll
    for (int off = 16; off > 0; off >>= 1) {
      frx += __shfl_xor(frx, off, 32);
      fry += __shfl_xor(fry, off, 32);
    }
    const float fx = (gx - cx) + 0.5f * (px - cx) + frx;
    const float fy = (gy - cy) + 0.5f * (py - cy) + fry;
    cx = fmaf(fx, DT_, cx);
    cy = fmaf(fy, DT_, cy);
    px = cx; py = cy;          // carry = (nxt, nxt)
    pre_x[t] = cx; pre_y[t] = cy;
  }

  // ---- speed-clipping scan (replicated; lane 0 stages results) ----
  const float min_sp = 4.087f * 0.85f;
  const float max_sp = 4.087f * 1.15f;
  cx = c0x; cy = c0y;
  #pragma unroll
  for (int t = 0; t < TSTEPS; ++t) {
    const float dx = pre_x[t] - cx, dy = pre_y[t] - cy;
    const float sp = sqrtf(fmaf(dx, dx, dy * dy));
    float clipped = fminf(fmaxf(sp, min_sp), max_sp);
    const float spf = (sp > 0.0f) ? clipped : sp;
    const float scale = spf / (sp + 1e-8f);
    cx = fmaf(dx, scale, cx);
    cy = fmaf(dy, scale, cy);
    if (lane == 0) {
      out_stage[wave][2 * t + 0] = cx;
      out_stage[wave][2 * t + 1] = cy;
    }
  }

  __syncthreads();
  // one coalesced 24-float store per agent: out[agent*24 + lane]
  if (lane < TSTEPS * 2) {
    out[(size_t)agent * (TSTEPS * 2) + lane] = out_stage[wave][lane];
  }
}

extern "C" void kernel_launch(void* const* d_in, const int* in_sizes, int n_in,
                              void* d_out, int out_size, void* d_ws, size_t ws_size,
                              hipStream_t stream) {
  (void)in_sizes; (void)n_in; (void)out_size; (void)d_ws; (void)ws_size;
  const float* history   = (const float*)d_in[0];
  const float* neighbors = (const float*)d_in[1];
  const float* goal      = (const float*)d_in[2];
  float* out = (float*)d_out;

  const int blocks = AGENTS / WAVES_PER_BLOCK;  // 4096
  pfm_rollout_kernel<<<blocks, 256, 0, stream>>>(history, neighbors, goal, out);
}